// Interp1D_39522289058146
// MI455X (gfx1250) — compile-verified
//
#include <hip/hip_runtime.h>

// ---------------------------------------------------------------------------
// Interp1D on MI455X (gfx1250):
//   - bandwidth-bound: ~36 MB HBM traffic -> ~1.6us floor at 23.3 TB/s
//   - y table (8x4096 f32 = 128 KB) staged into LDS per workgroup with 16
//     straight-line global_load_async_to_lds_b128 (ASYNCcnt + s_wait_asynccnt)
//   - 4 queries/thread vectorized: b128 loads of x_new, b128 stores of out
//   - r = yf + (yc-yf)*frac: exact when ce==fl (diff==0), no per-b select
//   - t uses a correctly-rounded divide so fl/ce/neq match jnp bit-exactly
// ---------------------------------------------------------------------------

#define N_GRID 4096
#define B_DIM  8
#define BLOCK  512                      // 16 waves (wave32)
#define VEC    4                        // queries per thread per group (float4)
#define GROUPS 2                        // 8 queries per thread total
#define QPB    (BLOCK * VEC * GROUPS)   // 4096 queries per block
#define TOT4   ((B_DIM * N_GRID) / 4)   // 8192 float4 tiles of y
#define COPY_ITERS (TOT4 / BLOCK)       // 16 async b128 per thread

#if defined(__has_builtin)
#  if __has_builtin(__builtin_amdgcn_global_load_async_to_lds_b128)
#    define HAVE_ASYNC_COPY 1
#  endif
#  if __has_builtin(__builtin_amdgcn_s_wait_asynccnt)
#    define HAVE_WAIT_ASYNC 1
#  endif
#  if __has_builtin(__builtin_amdgcn_rcpf)
#    define RCPF(x) __builtin_amdgcn_rcpf(x)
#  else
#    define RCPF(x) (1.0f / (x))
#  endif
#else
#  define RCPF(x) (1.0f / (x))
#endif

// Builtin parameter types (probe-confirmed by round-1 diagnostic):
//   param0: vector_size(16) int, AS1 (global) ; param1: same in AS3 (LDS)
typedef int v4i __attribute__((vector_size(16)));
typedef __attribute__((address_space(1))) v4i Gv4i;
typedef __attribute__((address_space(3))) v4i Lv4i;

__device__ __forceinline__ int clampi(int v, int lo, int hi) {
    return v < lo ? lo : (v > hi ? hi : v);
}

__global__ __launch_bounds__(BLOCK) void Interp1D_kernel(
    const float* __restrict__ x,     // (1, N)  uniform grid
    const float* __restrict__ y,     // (B, N)  table
    const float* __restrict__ xn,    // (Q,)    query points
    float* __restrict__ out,         // (B, Q)
    int Q)
{
    extern __shared__ float ylds[];  // B_DIM * N_GRID floats = 128 KB

    const int tid = threadIdx.x;

    // ---- Stage y into LDS: 16 unconditional async b128 per thread -------
    {
        const float4* src4 = (const float4*)y;
        float4*       dst4 = (float4*)ylds;
#if defined(HAVE_ASYNC_COPY)
#pragma unroll
        for (int k = 0; k < COPY_ITERS; ++k) {
            const int i = tid + k * BLOCK;
            __builtin_amdgcn_global_load_async_to_lds_b128(
                (Gv4i*)(src4 + i), (Lv4i*)(dst4 + i), /*offset*/0, /*cpol*/0);
        }
#  if defined(HAVE_WAIT_ASYNC)
        __builtin_amdgcn_s_wait_asynccnt(0);
#  else
        asm volatile("s_wait_asynccnt 0" ::: "memory");
#  endif
#else
#pragma unroll
        for (int k = 0; k < COPY_ITERS; ++k) {
            const int i = tid + k * BLOCK;
            dst4[i] = src4[i];
        }
#endif
    }
    __syncthreads();

    const float x0  = x[0];
    const float dxv = __fsub_rn(x[1], x[0]);

#pragma unroll
    for (int g = 0; g < GROUPS; ++g) {
        const int q0 = blockIdx.x * QPB + g * (BLOCK * VEC) + tid * VEC;

        if (q0 + VEC <= Q) {
            // ---------------- fast vectorized path ----------------------
            const float4 xq4 = *(const float4*)(xn + q0);
            const float xqa[VEC] = {xq4.x, xq4.y, xq4.z, xq4.w};

            int   idxF[VEC], idxC[VEC];
            float frac[VEC];
#pragma unroll
            for (int j = 0; j < VEC; ++j) {
                const float xq = xqa[j];
                // correctly-rounded divide: fl/ce/neq match jnp bit-exactly
                const float t = __fdiv_rn(__fsub_rn(xq, x0), dxv);
                int fl = clampi((int)floorf(t), 0, N_GRID - 1);
                int ce = clampi((int)ceilf(t),  0, N_GRID - 1);
                const float xf = x0 + dxv * (float)fl;
                const float xc = x0 + dxv * (float)ce;
                const float den = (ce != fl) ? (xc - xf) : 1.0f;
                frac[j] = (xq - xf) * RCPF(den);
                idxF[j] = fl;
                idxC[j] = ce;
            }

#pragma unroll
            for (int b = 0; b < B_DIM; ++b) {
                float oa[VEC];
#pragma unroll
                for (int j = 0; j < VEC; ++j) {
                    const float yf = ylds[b * N_GRID + idxF[j]];
                    const float yc = ylds[b * N_GRID + idxC[j]];
                    // exact (== yc == yf) when ce==fl since yc-yf == 0
                    oa[j] = __builtin_fmaf(yc - yf, frac[j], yf);
                }
                *(float4*)(out + (size_t)b * Q + q0) =
                    make_float4(oa[0], oa[1], oa[2], oa[3]);
            }
        } else {
            // ---------------- scalar guarded tail ------------------------
            for (int j = 0; j < VEC; ++j) {
                const int q = q0 + j;
                if (q >= Q) break;
                const float xq = xn[q];
                const float t  = __fdiv_rn(__fsub_rn(xq, x0), dxv);
                const int fl = clampi((int)floorf(t), 0, N_GRID - 1);
                const int ce = clampi((int)ceilf(t),  0, N_GRID - 1);
                const float xf = x0 + dxv * (float)fl;
                const float xc = x0 + dxv * (float)ce;
                const float den = (ce != fl) ? (xc - xf) : 1.0f;
                const float fr  = (xq - xf) * RCPF(den);
                for (int b = 0; b < B_DIM; ++b) {
                    const float yf = ylds[b * N_GRID + fl];
                    const float yc = ylds[b * N_GRID + ce];
                    out[(size_t)b * Q + q] = __builtin_fmaf(yc - yf, fr, yf);
                }
            }
        }
    }
}

extern "C" void kernel_launch(void* const* d_in, const int* in_sizes, int n_in,
                              void* d_out, int out_size, void* d_ws, size_t ws_size,
                              hipStream_t stream) {
    const float* x  = (const float*)d_in[0];   // 4096
    const float* y  = (const float*)d_in[1];   // 8*4096
    const float* xn = (const float*)d_in[2];   // 1,048,576
    float* out = (float*)d_out;                // 8 * 1,048,576

    const int Q = in_sizes[2];
    const int nblocks = (Q + QPB - 1) / QPB;   // 256 for the reference shape
    const size_t lds_bytes = (size_t)B_DIM * N_GRID * sizeof(float); // 128 KB

    hipLaunchKernelGGL(Interp1D_kernel, dim3(nblocks), dim3(BLOCK), lds_bytes,
                       stream, x, y, xn, out, Q);
}